// pagerank_explain_35656818491452
// MI455X (gfx1250) — compile-verified
//
#include <hip/hip_runtime.h>
#include <hip/hip_bf16.h>

#define GENES 20000
#define BATCH 256
#define NNZE  640000
#define NCLS  33
#define ALPHA 0.1f
#define H1DIM 1024
#define H2DIM 128

typedef __attribute__((ext_vector_type(2))) float v2f;
typedef __attribute__((ext_vector_type(8))) float v8f;

// ---------------------------------------------------------------------------
// CDNA5 async global->LDS DMA (ASYNCcnt-tracked), ISA 15.18.3 op 98.
// lds_off: per-lane LDS byte address; gaddr: per-lane 64-bit global address.
// ---------------------------------------------------------------------------
__device__ __forceinline__ void async_copy_b128(unsigned lds_off, const float* gaddr) {
    asm volatile("global_load_async_to_lds_b128 %0, %1, off"
                 :: "v"(lds_off), "v"((unsigned long long)(size_t)gaddr)
                 : "memory");
}
__device__ __forceinline__ void wait_async0() {
    asm volatile("s_wait_asynccnt 0x0" ::: "memory");
}

// ---------------------------------------------------------------------------
// Kernel 1: scalar-affine embed + relu + transpose [B,G] -> [G,B]
// ---------------------------------------------------------------------------
__global__ __launch_bounds__(256)
void embed_transpose_kernel(const float* __restrict__ xsamp,
                            const float* __restrict__ xtf,
                            const float* __restrict__ em_, const float* __restrict__ bm_,
                            const float* __restrict__ ee_, const float* __restrict__ be_,
                            float* __restrict__ xs0, float* __restrict__ xt0) {
    __shared__ float ts[32][33];
    __shared__ float tt[32][33];
    const int gt = blockIdx.x * 32;
    const int bt = blockIdx.y * 32;
    const int tx = threadIdx.x;   // 0..31
    const int ty = threadIdx.y;   // 0..7
    const float em = em_[0], bm = bm_[0], ee = ee_[0], be = be_[0];
    #pragma unroll
    for (int i = 0; i < 32; i += 8) {
        const int b = bt + ty + i;
        const int g = gt + tx;                    // coalesced in g
        const size_t src = (size_t)b * GENES + g;
        ts[ty + i][tx] = fmaxf(em * xsamp[src] + bm, 0.0f);
        tt[ty + i][tx] = fmaxf(ee * xtf[src]   + be, 0.0f);
    }
    __syncthreads();
    #pragma unroll
    for (int i = 0; i < 32; i += 8) {
        const int g = gt + ty + i;
        const int b = bt + tx;                    // coalesced in b
        xs0[(size_t)g * BATCH + b] = ts[tx][ty + i];
        xt0[(size_t)g * BATCH + b] = tt[tx][ty + i];
    }
}

// ---------------------------------------------------------------------------
// Kernel 2: COO SpMM via f32 atomics: y[r,:] += scale*val*x[c,:]
// one thread per (edge, 4-column chunk); x,y are [G,BATCH] row-major.
// x (20.5 MB) lives in L2, so gathers/atomics are L2-bandwidth bound.
// ---------------------------------------------------------------------------
__global__ __launch_bounds__(256)
void spmm_atomic_kernel(const int* __restrict__ rows, const int* __restrict__ cols,
                        const float* __restrict__ vals,
                        const float* __restrict__ x, float* __restrict__ y,
                        float scale) {
    const unsigned tid = blockIdx.x * blockDim.x + threadIdx.x;
    const unsigned total = (unsigned)NNZE * (BATCH / 4);
    if (tid >= total) return;
    const unsigned e  = tid >> 6;    // 64 chunks per edge
    const unsigned cb = tid & 63;
    const int r = rows[e];
    const int c = cols[e];
    const float v = vals[e] * scale;
    const float4 xv = ((const float4*)(x + (size_t)c * BATCH))[cb];
    float* yr = y + (size_t)r * BATCH + cb * 4;
    atomicAdd(yr + 0, v * xv.x);
    atomicAdd(yr + 1, v * xv.y);
    atomicAdd(yr + 2, v * xv.z);
    atomicAdd(yr + 3, v * xv.w);
}

// ---------------------------------------------------------------------------
// Kernel 3: PPR finalize + hadamard + transpose [G,B] -> [B,G]
// ---------------------------------------------------------------------------
__global__ __launch_bounds__(256)
void hadamard_transpose_kernel(const float* __restrict__ xs2, const float* __restrict__ xt2,
                               const float* __restrict__ xs0, const float* __restrict__ xt0,
                               float* __restrict__ had) {
    __shared__ float t[32][33];
    const int gt = blockIdx.x * 32;
    const int bt = blockIdx.y * 32;
    const int tx = threadIdx.x;
    const int ty = threadIdx.y;
    #pragma unroll
    for (int i = 0; i < 32; i += 8) {
        const int g = gt + ty + i;
        const int b = bt + tx;                    // coalesced in b
        const size_t idx = (size_t)g * BATCH + b;
        const float a = xs2[idx] + ALPHA * xs0[idx];
        const float c = xt2[idx] + ALPHA * xt0[idx];
        t[ty + i][tx] = a * c;                    // t[g_local][b_local]
    }
    __syncthreads();
    #pragma unroll
    for (int i = 0; i < 32; i += 8) {
        const int b = bt + ty + i;
        const int g = gt + tx;                    // coalesced in g
        had[(size_t)b * GENES + g] = t[tx][ty + i];
    }
}

// ---------------------------------------------------------------------------
// Kernel 4/5: f32 WMMA GEMM:  C[M,N] = relu( A[M,K] @ W[N,K]^T + bias )
// block tile 64x64, 8 waves, wave tile 16x32 (two 16x16 accumulators).
// K staged 32 at a time via double-buffered ASYNC global->LDS DMA, so the
// async engine prefetches slab i+1 while V_WMMA_F32_16X16X4_F32 consumes i.
// Requires: M%64==0, N%64==0, K%32==0 (true for all call sites here).
// ---------------------------------------------------------------------------
__global__ __launch_bounds__(256)
void gemm_wmma_f32_kernel(const float* __restrict__ A, const float* __restrict__ W,
                          const float* __restrict__ bias, float* __restrict__ C,
                          int K, int N, int applyRelu) {
    __shared__ __align__(16) float sA[2][64][36];   // 36-fl rows: 144B (16B-aligned)
    __shared__ __align__(16) float sB[2][64][36];

    const int tid  = threadIdx.x;
    const int wave = tid >> 5;          // 0..7
    const int lane = tid & 31;
    const int lh   = lane & 15;
    const int khal = (lane < 16) ? 0 : 2;

    const int m0 = blockIdx.x * 64;
    const int n0 = blockIdx.y * 64;
    const int wm = (wave >> 1) * 16;    // 0,16,32,48
    const int wn = (wave & 1) * 32;     // 0,32

    // cooperative fill: 64 rows x 32 cols, two b128 DMAs per thread per matrix
    const int lr = tid >> 2;            // 0..63
    const int lc = (tid & 3) * 4;       // 0,4,8,12  (also +16)

    const float* gA = A + (size_t)(m0 + lr) * K + lc;
    const float* gB = W + (size_t)(n0 + lr) * K + lc;

    v8f c00 = {};
    v8f c01 = {};

    // prefetch slab 0 into buffer 0
    async_copy_b128((unsigned)(size_t)&sA[0][lr][lc],      gA);
    async_copy_b128((unsigned)(size_t)&sA[0][lr][lc + 16], gA + 16);
    async_copy_b128((unsigned)(size_t)&sB[0][lr][lc],      gB);
    async_copy_b128((unsigned)(size_t)&sB[0][lr][lc + 16], gB + 16);

    int cur = 0;
    for (int k0 = 0; k0 < K; k0 += 32) {
        wait_async0();        // own DMAs into `cur` complete
        __syncthreads();      // all waves' DMAs visible; prior reads of `nxt` retired
        const int nxt = cur ^ 1;
        if (k0 + 32 < K) {    // uniform branch: EXEC stays all-ones for WMMA
            const float* pA = gA + k0 + 32;
            const float* pB = gB + k0 + 32;
            async_copy_b128((unsigned)(size_t)&sA[nxt][lr][lc],      pA);
            async_copy_b128((unsigned)(size_t)&sA[nxt][lr][lc + 16], pA + 16);
            async_copy_b128((unsigned)(size_t)&sB[nxt][lr][lc],      pB);
            async_copy_b128((unsigned)(size_t)&sB[nxt][lr][lc + 16], pB + 16);
        }
        const float (*tAp)[36] = sA[cur];
        const float (*tBp)[36] = sB[cur];
        #pragma unroll
        for (int kk = 0; kk < 32; kk += 4) {
            const int kb = kk + khal;
            // A frag 16x4 (ISA 7.12.2): lanes 0-15 K={kk,kk+1}, lanes 16-31 K={kk+2,kk+3}
            v2f av;  av.x  = tAp[wm + lh][kb];       av.y  = tAp[wm + lh][kb + 1];
            // B frags 4x16: lane = column n, VGPR pair over K (mirrors A)
            v2f bv0; bv0.x = tBp[wn + lh][kb];       bv0.y = tBp[wn + lh][kb + 1];
            v2f bv1; bv1.x = tBp[wn + 16 + lh][kb];  bv1.y = tBp[wn + 16 + lh][kb + 1];
            c00 = __builtin_amdgcn_wmma_f32_16x16x4_f32(
                      false, av, false, bv0, (short)0, c00, false, false);
            c01 = __builtin_amdgcn_wmma_f32_16x16x4_f32(
                      false, av, false, bv1, (short)0, c01, false, false);
        }
        cur = nxt;
    }

    // C/D layout (ISA 7.12.2): VGPR r -> M = wm + r (+8 for lanes 16-31), N = lane&15
    const int cn0 = n0 + wn + lh;
    const float bb0 = bias[cn0];
    const float bb1 = bias[cn0 + 16];
    const int mrow = m0 + wm + ((lane < 16) ? 0 : 8);
    #pragma unroll
    for (int r = 0; r < 8; ++r) {
        float v0 = c00[r] + bb0;
        float v1 = c01[r] + bb1;
        if (applyRelu) { v0 = fmaxf(v0, 0.0f); v1 = fmaxf(v1, 0.0f); }
        C[(size_t)(mrow + r) * N + cn0]      = v0;
        C[(size_t)(mrow + r) * N + cn0 + 16] = v1;
    }
}

// ---------------------------------------------------------------------------
// Kernel 6: output layer  out[b,n] = h2[b,:] . W3[n,:] + b3[n]   (N=33)
// ---------------------------------------------------------------------------
__global__ __launch_bounds__(64)
void out_layer_kernel(const float* __restrict__ h2, const float* __restrict__ W3,
                      const float* __restrict__ b3, float* __restrict__ out) {
    const int b = blockIdx.x;
    const int n = threadIdx.x;
    if (n >= NCLS) return;
    float s = b3[n];
    const float* hr = h2 + (size_t)b * H2DIM;
    const float* wr = W3 + (size_t)n * H2DIM;
    #pragma unroll 8
    for (int k = 0; k < H2DIM; ++k) s += hr[k] * wr[k];
    out[(size_t)b * NCLS + n] = s;
}

// ---------------------------------------------------------------------------
extern "C" void kernel_launch(void* const* d_in, const int* in_sizes, int n_in,
                              void* d_out, int out_size, void* d_ws, size_t ws_size,
                              hipStream_t stream) {
    const float* x_sample  = (const float*)d_in[0];
    const float* x_TF      = (const float*)d_in[1];
    const int*   adj_rows  = (const int*)  d_in[2];
    const int*   adj_cols  = (const int*)  d_in[3];
    const float* adj_vals  = (const float*)d_in[4];
    const int*   adjT_rows = (const int*)  d_in[5];
    const int*   adjT_cols = (const int*)  d_in[6];
    const float* adjT_vals = (const float*)d_in[7];
    const float* emb_mut   = (const float*)d_in[8];
    const float* bias_mut  = (const float*)d_in[9];
    const float* emb_exp   = (const float*)d_in[10];
    const float* bias_exp  = (const float*)d_in[11];
    const float* W1        = (const float*)d_in[12];
    const float* b1        = (const float*)d_in[13];
    const float* W2        = (const float*)d_in[14];
    const float* b2        = (const float*)d_in[15];
    const float* W3        = (const float*)d_in[16];
    const float* b3        = (const float*)d_in[17];

    const size_t GB = (size_t)GENES * BATCH;
    float* ws  = (float*)d_ws;
    float* xs0 = ws;            // hs  [G,B]
    float* xt0 = ws + 1 * GB;   // ht  [G,B]
    float* xA  = ws + 2 * GB;   // ping
    float* tA  = ws + 3 * GB;
    float* xB  = ws + 4 * GB;   // pong
    float* tB  = ws + 5 * GB;
    float* had = ws + 6 * GB;   // [B,G]
    float* h1  = ws + 7 * GB;   // [B,1024]
    float* h2  = h1 + (size_t)BATCH * H1DIM;

    // 1) embed + transpose
    embed_transpose_kernel<<<dim3(GENES / 32, BATCH / 32), dim3(32, 8), 0, stream>>>(
        x_sample, x_TF, emb_mut, bias_mut, emb_exp, bias_exp, xs0, xt0);

    // 2) PPR iteration 1: x <- A @ (0.9*x)
    hipMemsetAsync(xA, 0, GB * sizeof(float), stream);
    hipMemsetAsync(tA, 0, GB * sizeof(float), stream);
    const unsigned spThreads = (unsigned)NNZE * (BATCH / 4);
    const unsigned spBlocks  = (spThreads + 255) / 256;
    spmm_atomic_kernel<<<spBlocks, 256, 0, stream>>>(adj_rows,  adj_cols,  adj_vals,  xs0, xA, 1.0f - ALPHA);
    spmm_atomic_kernel<<<spBlocks, 256, 0, stream>>>(adjT_rows, adjT_cols, adjT_vals, xt0, tA, 1.0f - ALPHA);

    // 3) PPR iteration 2
    hipMemsetAsync(xB, 0, GB * sizeof(float), stream);
    hipMemsetAsync(tB, 0, GB * sizeof(float), stream);
    spmm_atomic_kernel<<<spBlocks, 256, 0, stream>>>(adj_rows,  adj_cols,  adj_vals,  xA, xB, 1.0f - ALPHA);
    spmm_atomic_kernel<<<spBlocks, 256, 0, stream>>>(adjT_rows, adjT_cols, adjT_vals, tA, tB, 1.0f - ALPHA);

    // 4) finalize + hadamard + transpose to [B,G]
    hadamard_transpose_kernel<<<dim3(GENES / 32, BATCH / 32), dim3(32, 8), 0, stream>>>(
        xB, tB, xs0, xt0, had);

    // 5) h1 = relu(had @ W1^T + b1)   [256,20000]x[20000,1024]
    gemm_wmma_f32_kernel<<<dim3(BATCH / 64, H1DIM / 64), 256, 0, stream>>>(
        had, W1, b1, h1, GENES, H1DIM, 1);

    // 6) h2 = relu(h1 @ W2^T + b2)    [256,1024]x[1024,128]
    gemm_wmma_f32_kernel<<<dim3(BATCH / 64, H2DIM / 64), 256, 0, stream>>>(
        h1, W2, b2, h2, H1DIM, H2DIM, 1);

    // 7) out = h2 @ W3^T + b3         [256,128]x[128,33]
    out_layer_kernel<<<BATCH, 64, 0, stream>>>(h2, W3, b3, (float*)d_out);
}